// ProGraphConv_4157528342563
// MI455X (gfx1250) — compile-verified
//
#include <hip/hip_runtime.h>
#include <hip/hip_bf16.h>

#define NN 30000
#define PP 4
#define DD 64
#define FD 256            // P*D
#define EE 480000
#define MIN_NORM 1e-15f
#define EPSF 1e-7f
#define MAX_NORMF 1000.0f

typedef float v2f __attribute__((ext_vector_type(2)));
typedef float v8f __attribute__((ext_vector_type(8)));

// ---------------- wave32 helpers ----------------
__device__ __forceinline__ float wave_sum(float v) {
    v += __shfl_xor(v, 16, 32);
    v += __shfl_xor(v, 8, 32);
    v += __shfl_xor(v, 4, 32);
    v += __shfl_xor(v, 2, 32);
    v += __shfl_xor(v, 1, 32);
    return v;
}
__device__ __forceinline__ float bcast0(float v) { return __shfl(v, 0, 32); }
__device__ __forceinline__ float arcosh_f(float x) {
    return logf(x + sqrtf(fmaxf(x * x - 1.0f, MIN_NORM)));
}

// Each lane of a wave holds elements (2*lane, 2*lane+1) of a 64-vector.
// Element 0 is the "time" component.
struct R2 { float a, b; };

// proj: replace element 0 with sqrt(K + sum_{e>=1} v_e^2), K = 1
__device__ __forceinline__ R2 proj_row(R2 h, int lane) {
    float ysq = wave_sum(((lane == 0) ? 0.0f : h.a * h.a) + h.b * h.b);
    float t = sqrtf(fmaxf(1.0f + ysq, EPSF));
    if (lane == 0) h.a = t;
    return h;
}

// proj(expmap0(proj_tan0(v))), K = 1
__device__ __forceinline__ R2 exp0_row(R2 v, int lane) {
    float s0 = (lane == 0) ? 0.0f : v.a;
    float nsq = wave_sum(s0 * s0 + v.b * v.b);
    float n = fmaxf(sqrtf(nsq), MIN_NORM);
    float shn = sinhf(n) / n;
    R2 h;
    h.a = (lane == 0) ? coshf(n) : shn * s0;
    h.b = shn * v.b;
    return proj_row(h, lane);
}

// logmap0, K = 1
__device__ __forceinline__ R2 log0_row(R2 h, int lane) {
    float s0 = (lane == 0) ? 0.0f : h.a;
    float nsq = wave_sum(s0 * s0 + h.b * h.b);
    float n = fmaxf(sqrtf(nsq), MIN_NORM);
    float x0 = bcast0(h.a);
    float theta = fmaxf(x0, 1.0f + EPSF);
    float f = arcosh_f(theta) / n;
    R2 o;
    o.a = f * s0;          // lane0 -> 0
    o.b = f * h.b;
    return o;
}

// mobius_add(h, hb) = proj(expmap(ptransp0(h, logmap0(hb)), h)), K = 1
__device__ __forceinline__ R2 mobius_add_row(R2 h, R2 hb, int lane) {
    R2 ub = log0_row(hb, lane);                 // tangent at origin, ub[0]=0
    float x0 = bcast0(h.a);                     // time of h
    float y0 = (lane == 0) ? 0.0f : h.a;        // space of h
    float ynsq = wave_sum(y0 * y0 + h.b * h.b);
    float yn = fmaxf(sqrtf(ynsq), MIN_NORM);
    float yua = y0 / yn, yub = h.b / yn;
    float alpha = wave_sum(yua * ub.a + yub * ub.b);
    // w_space = ub_space - alpha*(1 - x0)*yu ; w_time discarded by proj_tan
    float c = alpha * (1.0f - x0);
    float wa = ub.a - c * yua;
    float wb = ub.b - c * yub;
    // proj_tan: t = (sum_{e>=1} h_e*w_e) / max(x0, MIN)
    float ux = wave_sum(y0 * wa + h.b * wb);
    float t = ux / fmaxf(x0, MIN_NORM);
    float u2a = (lane == 0) ? t : wa;
    float u2b = wb;
    // expmap at h: minkdot = sum_{e>=1} u2^2 - u2_0^2
    float sd = wave_sum(((lane == 0) ? 0.0f : u2a * u2a) + u2b * u2b);
    float md = sd - t * t;
    float normu = fminf(sqrtf(fmaxf(md, EPSF)), MAX_NORMF);
    float th = fmaxf(normu, MIN_NORM);
    float chh = coshf(th), shh = sinhf(th) / th;
    R2 r;
    r.a = chh * h.a + shh * u2a;
    r.b = chh * h.b + shh * u2b;
    return proj_row(r, lane);
}

// ---------------- kernel 1: u = logmap0(x) ----------------
__global__ __launch_bounds__(256) void k_log0(const float* __restrict__ X,
                                              float* __restrict__ U) {
    int row = blockIdx.x * 8 + (threadIdx.x >> 5);   // row = n*P + p
    int lane = threadIdx.x & 31;
    v2f m = *(const v2f*)(X + (size_t)row * DD + 2 * lane);
    R2 o = log0_row({m.x, m.y}, lane);
    v2f w; w.x = o.a; w.y = o.b;
    *(v2f*)(U + (size_t)row * DD + 2 * lane) = w;
}

// ---------------- kernel 2: mu = u @ W^T via V_WMMA_F32_16X16X4_F32 ------
// block = 512 threads (16 waves). Block computes 16 rows x 256 cols.
// Wave w computes output tile rows[i0..i0+15] x cols[16w..16w+15].
__global__ __launch_bounds__(512) void k_gemm(const float* __restrict__ U,
                                              const float* __restrict__ W,
                                              float* __restrict__ MU) {
    __shared__ float lds[16 * 260];                  // 16-row panel, pitch 260
    const int i0 = blockIdx.x * 16;
    const int t = threadIdx.x;
    {   // stage 16x256 of U, coalesced (each thread: 8 consecutive floats)
        int r = t >> 5;                // 0..15
        int c0 = (t & 31) * 8;         // 0..248
        const float* src = U + (size_t)(i0 + r) * FD + c0;
        float4 a = *(const float4*)(src);
        float4 b = *(const float4*)(src + 4);
        float* dst = &lds[r * 260 + c0];
        *(float4*)(dst) = a;
        *(float4*)(dst + 4) = b;
    }
    __syncthreads();

    const int wave = t >> 5;           // 0..15 -> column tile
    const int lane = t & 31;
    const int j0 = wave * 16;
    const int m = lane & 15;           // M (for A) / N (for B)
    const int khalf = (lane >> 4) * 2; // 0 or 2: K sub-offset per lane half

    v8f acc = {};
    const float* wbase = W + (size_t)(j0 + m) * FD + khalf;  // B[k][n] = W[j0+n][k]
    const float* abase = &lds[m * 260 + khalf];              // A[m][k] = u[i0+m][k]

#pragma unroll 8
    for (int k0 = 0; k0 < FD; k0 += 4) {
        v2f a = *(const v2f*)(abase + k0);   // {A[m][k0+kh], A[m][k0+kh+1]}
        v2f b = *(const v2f*)(wbase + k0);   // {B[k0+kh][n], B[k0+kh+1][n]}
        acc = __builtin_amdgcn_wmma_f32_16x16x4_f32(
            false, a, false, b, (short)0, acc, false, false);
    }

    // C/D layout: VGPR v -> row i0 + v + 8*(lane>>4), col j0 + (lane&15)
    const int rbase = i0 + ((lane >> 4) << 3);
    const int col = j0 + m;
#pragma unroll
    for (int v = 0; v < 8; ++v) {
        MU[(size_t)(rbase + v) * FD + col] = acc[v];
    }
}

// ---------------- kernel 3: mu -> xt (expmap0/proj/mobius/logmap0) -------
__global__ __launch_bounds__(256) void k_stageA(const float* __restrict__ MU,
                                                const float* __restrict__ bias,
                                                float* __restrict__ XT) {
    int row = blockIdx.x * 8 + (threadIdx.x >> 5);   // row = n*P + p
    int lane = threadIdx.x & 31;
    int p = row & (PP - 1);
    v2f m = *(const v2f*)(MU + (size_t)row * DD + 2 * lane);
    R2 h = exp0_row({m.x, m.y}, lane);
    v2f bv = *(const v2f*)(bias + p * DD + 2 * lane);
    R2 hb = exp0_row({bv.x, bv.y}, lane);
    h = mobius_add_row(h, hb, lane);
    h = proj_row(h, lane);
    R2 xt = log0_row(h, lane);
    v2f o; o.x = xt.a; o.y = xt.b;
    *(v2f*)(XT + (size_t)row * DD + 2 * lane) = o;
}

// ---------------- kernel 4: zero support ---------------------------------
__global__ __launch_bounds__(256) void k_zero(float4* __restrict__ S) {
    S[blockIdx.x * 256 + threadIdx.x] = float4{0.f, 0.f, 0.f, 0.f};
}

// ---------------- kernel 5: SpMM (edge-parallel, atomic scatter) ---------
__global__ __launch_bounds__(256) void k_spmm(const float* __restrict__ XT,
                                              const float* __restrict__ val,
                                              const int* __restrict__ colI,
                                              const int* __restrict__ rowI,
                                              float* __restrict__ SUP) {
    int e = blockIdx.x * 8 + (threadIdx.x >> 5);
    if (e >= EE) return;
    int lane = threadIdx.x & 31;
    int c = colI[e];
    int r = rowI[e];
    float v = val[e];
    const float* src = XT + (size_t)c * FD + lane * 8;
    float* dst = SUP + (size_t)r * FD + lane * 8;
    float4 a = *(const float4*)(src);
    float4 b = *(const float4*)(src + 4);
    atomicAdd(dst + 0, v * a.x);
    atomicAdd(dst + 1, v * a.y);
    atomicAdd(dst + 2, v * a.z);
    atomicAdd(dst + 3, v * a.w);
    atomicAdd(dst + 4, v * b.x);
    atomicAdd(dst + 5, v * b.y);
    atomicAdd(dst + 6, v * b.z);
    atomicAdd(dst + 7, v * b.w);
}

// ---------------- kernel 6: support -> out -------------------------------
__global__ __launch_bounds__(256) void k_stageC(const float* __restrict__ SUP,
                                                float* __restrict__ OUT) {
    int row = blockIdx.x * 8 + (threadIdx.x >> 5);
    int lane = threadIdx.x & 31;
    v2f s = *(const v2f*)(SUP + (size_t)row * DD + 2 * lane);
    s.x = fminf(s.x, MAX_NORMF);
    s.y = fminf(s.y, MAX_NORMF);
    R2 h = exp0_row({s.x, s.y}, lane);
    R2 t = log0_row(h, lane);
    t.a = fminf(fmaxf(t.a, 0.0f), MAX_NORMF);
    t.b = fminf(fmaxf(t.b, 0.0f), MAX_NORMF);
    // out = expmap0(proj_tan0(t), Ko=1), no proj
    float s0 = (lane == 0) ? 0.0f : t.a;
    float n = fmaxf(sqrtf(wave_sum(s0 * s0 + t.b * t.b)), MIN_NORM);
    float shn = sinhf(n) / n;
    v2f o;
    o.x = (lane == 0) ? coshf(n) : shn * s0;
    o.y = shn * t.b;
    *(v2f*)(OUT + (size_t)row * DD + 2 * lane) = o;
}

// ---------------- driver -------------------------------------------------
extern "C" void kernel_launch(void* const* d_in, const int* in_sizes, int n_in,
                              void* d_out, int out_size, void* d_ws, size_t ws_size,
                              hipStream_t stream) {
    const float* x       = (const float*)d_in[0];
    const float* adj_val = (const float*)d_in[1];
    const float* weight  = (const float*)d_in[2];
    const float* bias    = (const float*)d_in[3];
    const int*   adj_row = (const int*)d_in[4];
    const int*   adj_col = (const int*)d_in[5];
    float* out = (float*)d_out;

    float* buf0 = (float*)d_ws;                    // u, then xt
    float* buf1 = buf0 + (size_t)NN * FD;          // mu, then support

    const int rowBlocks = (NN * PP) / 8;           // 15000 (wave per 64-row)

    // 1) u = logmap0(x)
    k_log0<<<rowBlocks, 256, 0, stream>>>(x, buf0);
    // 2) mu = u @ W^T   (WMMA f32)
    k_gemm<<<NN / 16, 512, 0, stream>>>(buf0, weight, buf1);
    // 3) xt = logmap0(proj(mobius_add(proj(expmap0(mu)), hyp_bias)))
    k_stageA<<<rowBlocks, 256, 0, stream>>>(buf1, bias, buf0);
    // 4) support = 0
    k_zero<<<(NN * FD / 4) / 256, 256, 0, stream>>>((float4*)buf1);
    // 5) support += adj_val * xt[col]  (scatter to row)
    k_spmm<<<EE / 8, 256, 0, stream>>>(buf0, adj_val, adj_col, adj_row, buf1);
    // 6) out = expmap0(relu(logmap0(proj(expmap0(min(support,MAX))))))
    k_stageC<<<rowBlocks, 256, 0, stream>>>(buf1, out);
}